// WaveletFeatures_77214922047612
// MI455X (gfx1250) — compile-verified
//
#include <hip/hip_runtime.h>

#ifndef __has_builtin
#define __has_builtin(x) 0
#endif

#if __has_builtin(__builtin_amdgcn_wmma_f32_16x16x4_f32)
#define HAVE_WMMA_F32_16X16X4 1
#else
#define HAVE_WMMA_F32_16X16X4 0
#endif

#if __has_builtin(__builtin_amdgcn_global_load_async_to_lds_b128)
#define HAVE_ASYNC_LDS 1
#else
#define HAVE_ASYNC_LDS 0
#endif

typedef float v2f __attribute__((ext_vector_type(2)));
typedef float v8f __attribute__((ext_vector_type(8)));
// Matches the async builtin's parameter type: 16-byte int vector.
typedef int v4i_b128 __attribute__((vector_size(4 * sizeof(int))));
#define AS_GLOBAL __attribute__((address_space(1)))
#define AS_LDS    __attribute__((address_space(3)))

// db4 decomposition filters, pre-reversed for the cross-correlation form:
//   y[o] = sum_{i=0..7} P[2o+1+i] * FREV[i],   FREV[i] = F[7-i]
// where P is the symmetric-padded signal (pywt 'symmetric' mode).
__device__ constexpr float LREV[8] = {
    0.23037781330885523f,  0.7148465705525415f,  0.6308807679295904f,
   -0.02798376941698385f, -0.18703481171888114f, 0.030841381835986965f,
    0.032883011666982945f, -0.010597401784997278f};
__device__ constexpr float HREV[8] = {
   -0.010597401784997278f, -0.032883011666982945f, 0.030841381835986965f,
    0.18703481171888114f,  -0.02798376941698385f, -0.6308807679295904f,
    0.7148465705525415f,   -0.23037781330885523f};

// One DWT level. P = padded signal in LDS (valid indices 1 .. 2M+6).
// Lanes 0-15 of each wave produce cA[o] (WMMA D row M=0), lanes 16-31
// produce cD[o] (D row M=8), o = it*128 + wave*16 + (lane&15).
__device__ __forceinline__ void dwt_level(
    const float* __restrict__ P, int M,
    float* __restrict__ aLds, float* __restrict__ aGlb,
    float* __restrict__ dGlb,
    v2f a1, v2f a2, int m, int lh, int waveId)
{
  const int iters = (M + 127) >> 7;   // 8 waves * 16 outputs each
  for (int it = 0; it < iters; ++it) {
    const int o = (it << 7) + (waveId << 4) + m;
#if HAVE_WMMA_F32_16X16X4
    // B (4x16): lane n<16 holds K=0,1 in v0,v1; lane n>=16 holds K=2,3.
    const int g = 2 * o + 1 + (lh << 1);
    v2f b1, b2;
    b1.x = P[g];     b1.y = P[g + 1];     // taps kb, kb+1
    b2.x = P[g + 4]; b2.y = P[g + 5];     // taps kb+4, kb+5
    v8f c = {};
    c = __builtin_amdgcn_wmma_f32_16x16x4_f32(false, a1, false, b1,
                                              (short)0, c, false, false);
    c = __builtin_amdgcn_wmma_f32_16x16x4_f32(false, a2, false, b2,
                                              (short)0, c, false, false);
    const float v = c[0];   // VGPR0: M=0 (lanes 0-15) = cA, M=8 (16-31) = cD
#else
    (void)a1; (void)a2;
    const int g = 2 * o + 1;
    float v = 0.f;
#pragma unroll
    for (int i = 0; i < 8; ++i)
      v += P[g + i] * (lh ? HREV[i] : LREV[i]);
#endif
    if (o < M) {
      if (lh == 0) {
        if (aLds) aLds[o] = v; else aGlb[o] = v;
      } else {
        dGlb[o] = v;
      }
    }
  }
}

__global__ __launch_bounds__(256) void dwt3_db4_kernel(
    const float* __restrict__ x, float* __restrict__ out, int rows)
{
  constexpr int N0 = 8192, M1 = 4099, M2 = 2053, M3 = 1030;
  // Padded lengths: N0+14=8206, M1+14=4113, M2+14=2067. x data placed at
  // lds[8] so async b128 writes are 16B-aligned. Slack floats at the end
  // absorb masked-tail reads (max touched idx 14630).
  __shared__ float lds[14656];
  float* P0 = lds + 1;       // padded input  (len 8206)
  float* P1 = lds + 8207;    // padded cA1    (len 4113)
  float* P2 = lds + 12320;   // padded cA2    (len 2067)

  const int row    = blockIdx.x;
  const int tid    = threadIdx.x;
  const int lane   = tid & 31;            // wave32
  const int waveId = tid >> 5;            // 8 waves
  const int m      = lane & 15;
  const int lh     = lane >> 4;           // 0: cA half, 1: cD half

  const float* xrow = x + (size_t)row * N0;
  float* xdst = P0 + 7;                   // = lds + 8, 32B aligned

  // ---- bulk row copy: global -> LDS (async-to-LDS, 16B per lane) ----
#if HAVE_ASYNC_LDS
#pragma unroll
  for (int it = 0; it < 8; ++it) {
    const int cidx = it * 256 + tid;      // 2048 chunks of 16B = 32KB
    __builtin_amdgcn_global_load_async_to_lds_b128(
        (AS_GLOBAL v4i_b128*)(xrow + 4 * cidx),
        (AS_LDS v4i_b128*)(xdst + 4 * cidx),
        0, 0);
  }
#else
#pragma unroll
  for (int it = 0; it < 8; ++it) {
    const int cidx = it * 256 + tid;
    *(float4*)(xdst + 4 * cidx) = *(const float4*)(xrow + 4 * cidx);
  }
#endif

  // symmetric edge padding of level-0 signal (disjoint LDS range vs async)
  if (tid < 7) {
    P0[tid] = xrow[6 - tid];
  } else if (tid < 14) {
    const int i = tid - 7;
    P0[7 + N0 + i] = xrow[N0 - 1 - i];
  }

#if HAVE_ASYNC_LDS
#if __has_builtin(__builtin_amdgcn_s_wait_asynccnt)
  __builtin_amdgcn_s_wait_asynccnt(0);
#else
  asm volatile("s_wait_asynccnt 0" ::: "memory");
#endif
#endif
  __syncthreads();

  // ---- per-wave constant A matrices (16x4 f32): row 0 = LO, row 8 = HI ----
  // Lanes 0-15 hold K=0,1 in v0,v1; lanes 16-31 hold K=2,3.
  v2f a1, a2;
  a1.x = (m == 0) ? (lh ? LREV[2] : LREV[0])
       : (m == 8) ? (lh ? HREV[2] : HREV[0]) : 0.f;
  a1.y = (m == 0) ? (lh ? LREV[3] : LREV[1])
       : (m == 8) ? (lh ? HREV[3] : HREV[1]) : 0.f;
  a2.x = (m == 0) ? (lh ? LREV[6] : LREV[4])
       : (m == 8) ? (lh ? HREV[6] : HREV[4]) : 0.f;
  a2.y = (m == 0) ? (lh ? LREV[7] : LREV[5])
       : (m == 8) ? (lh ? HREV[7] : HREV[5]) : 0.f;

  // ---- output segment pointers: [cA3 | cD3 | cD2 | cD1], each row-major ----
  const size_t R = (size_t)rows;
  float* outA3 = out + (size_t)row * M3;
  float* outD3 = out + R * M3            + (size_t)row * M3;
  float* outD2 = out + R * (2 * M3)      + (size_t)row * M2;
  float* outD1 = out + R * (2 * M3 + M2) + (size_t)row * M1;

  // ---- level 1: P0 -> cA1 (LDS), cD1 (global) ----
  dwt_level(P0, M1, P1 + 7, nullptr, outD1, a1, a2, m, lh, waveId);
  __syncthreads();
  if (tid < 7) {
    P1[tid] = P1[13 - tid];                       // a1[6-t]
  } else if (tid < 14) {
    const int i = tid - 7;
    P1[7 + M1 + i] = P1[7 + M1 - 1 - i];          // a1[M1-1-i]
  }
  __syncthreads();

  // ---- level 2: P1 -> cA2 (LDS), cD2 (global) ----
  dwt_level(P1, M2, P2 + 7, nullptr, outD2, a1, a2, m, lh, waveId);
  __syncthreads();
  if (tid < 7) {
    P2[tid] = P2[13 - tid];
  } else if (tid < 14) {
    const int i = tid - 7;
    P2[7 + M2 + i] = P2[7 + M2 - 1 - i];
  }
  __syncthreads();

  // ---- level 3: P2 -> cA3 (global), cD3 (global) ----
  dwt_level(P2, M3, nullptr, outA3, outD3, a1, a2, m, lh, waveId);
}

extern "C" void kernel_launch(void* const* d_in, const int* in_sizes, int n_in,
                              void* d_out, int out_size, void* d_ws, size_t ws_size,
                              hipStream_t stream) {
  (void)n_in; (void)out_size; (void)d_ws; (void)ws_size;
  const float* x = (const float*)d_in[0];
  float* out = (float*)d_out;
  const int rows = in_sizes[0] / 8192;   // 4096 for the reference shapes
  dwt3_db4_kernel<<<dim3(rows), dim3(256), 0, stream>>>(x, out, rows);
}